// GINEModel_79439715107083
// MI455X (gfx1250) — compile-verified
//
#include <hip/hip_runtime.h>
#include <hip/hip_bf16.h>

typedef float v2f __attribute__((ext_vector_type(2)));
typedef float v8f __attribute__((ext_vector_type(8)));

#define DCH 64   // feature dim
#define EDIM 16  // edge attr dim

// ---------------------------------------------------------------------------
// CDNA5 async copy: 16B per lane, global -> LDS, tracked by ASYNCcnt.
// LDS destination address is passed as a raw 32-bit LDS offset in a VGPR
// (generic -> addrspace(3) cast yields the offset), global source as a
// 64-bit VGPR pair.  GV addressing mode: "vdst, vaddr, off".
// ---------------------------------------------------------------------------
__device__ __forceinline__ void async_copy16(float* lds_dst, const float* gsrc) {
    unsigned loff = (unsigned)(size_t)(__attribute__((address_space(3))) float*)lds_dst;
    asm volatile("global_load_async_to_lds_b128 %0, %1, off"
                 :: "v"(loff), "v"(gsrc)
                 : "memory");
}
__device__ __forceinline__ void wait_asynccnt0() {
    asm volatile("s_wait_asynccnt 0x0" ::: "memory");
}

// ---------------------------------------------------------------------------
// Edge kernel: for a tile of 16 edges, e = edge_attr @ ew.T + eb via
// v_wmma_f32_16x16x4_f32 ; m = relu(x[src] + e) ; atomic scatter-add into agg.
// C/D layout: VGPR r -> row M (edge r in lanes 0-15, edge 8+r in lanes 16-31),
// lane&15 -> column N (output channel within the 16-wide N tile).
// ---------------------------------------------------------------------------
__global__ __launch_bounds__(256) void gine_edge_kernel(
    const float* __restrict__ x,      // [N,64] node features (x or h1)
    const float* __restrict__ eattr,  // [E,16]
    const int*   __restrict__ src,    // [E]
    const int*   __restrict__ dst,    // [E]
    const float* __restrict__ ew,     // [64,16] (out,in)
    const float* __restrict__ eb,     // [64]
    float*       __restrict__ agg,    // [N,64] pre-zeroed
    int nEdges)
{
    const int lane = threadIdx.x & 31;
    const int half = lane >> 4;         // 0: lanes 0-15, 1: lanes 16-31
    const int l15  = lane & 15;
    const int waveId = (blockIdx.x * (blockDim.x >> 5)) + (threadIdx.x >> 5);
    const int nWaves = (gridDim.x * blockDim.x) >> 5;

    // B = ew.T : B[k][n] = ew[n][k].  B VGPR layout for 16x16x4 f32:
    // vgpr0 lane -> {K = ks*4 + half*2, N = l15}, vgpr1 -> K+1  => one v2f/lane.
    v2f B[4][4];
    float bias[4];
#pragma unroll
    for (int nt = 0; nt < 4; ++nt) {
        const int n = nt * 16 + l15;
        bias[nt] = eb[n];
#pragma unroll
        for (int ks = 0; ks < 4; ++ks) {
            const int k = ks * 4 + half * 2;
            B[nt][ks] = *(const v2f*)(ew + n * EDIM + k);
        }
    }

    const int nTiles = nEdges >> 4;     // E = 1.6M -> 100000 tiles exactly
    for (int tile = waveId; tile < nTiles; tile += nWaves) {
        const int ebase = tile * 16;

        // A tile: A[M=l15][K], lanes 0-15 hold K = ks*4+{0,1}, lanes 16-31 K+2.
        v2f A[4];
        const float* arow = eattr + (ebase + l15) * EDIM + half * 2;
#pragma unroll
        for (int ks = 0; ks < 4; ++ks) A[ks] = *(const v2f*)(arow + ks * 4);

        if (tile + nWaves < nTiles) {   // wave-uniform -> EXEC stays full
            __builtin_prefetch(eattr + (ebase + nWaves * 16 + l15) * EDIM, 0, 1);
        }

        // src/dst of edge (ebase + l15) in lane l15 (replicated in both halves).
        // Broadcast to C-layout rows once; reuse across all 4 N-tiles.
        const int m_src = src[ebase + l15];
        const int m_dst = dst[ebase + l15];
        int sOff[8], tOff[8];           // flat elem offset of (edge row, ch l15)
#pragma unroll
        for (int r = 0; r < 8; ++r) {
            sOff[r] = __shfl(m_src, r + half * 8, 32) * DCH + l15;
            tOff[r] = __shfl(m_dst, r + half * 8, 32) * DCH + l15;
        }

#pragma unroll
        for (int nt = 0; nt < 4; ++nt) {
            v8f c;
#pragma unroll
            for (int i = 0; i < 8; ++i) c[i] = bias[nt];
#pragma unroll
            for (int ks = 0; ks < 4; ++ks)
                c = __builtin_amdgcn_wmma_f32_16x16x4_f32(
                        false, A[ks], false, B[nt][ks], (short)0, c, false, false);

            const int coff = nt * 16;
            // batch the 8 gathers so the loads overlap, then relu+atomic burst
            float xv[8];
#pragma unroll
            for (int r = 0; r < 8; ++r) xv[r] = x[sOff[r] + coff];
#pragma unroll
            for (int r = 0; r < 8; ++r) {
                float m = c[r] + xv[r];             // relu(x_j + e)
                m = m > 0.f ? m : 0.f;
                atomicAdd(&agg[tOff[r] + coff], m); // global_atomic_add_f32
            }
        }
    }
}

// ---------------------------------------------------------------------------
// Node kernel: h_out = relu((xin+agg) @ w1.T + b1) @ w2.T + b2 for 16 nodes
// per wave.  Weights staged in LDS via global_load_async_to_lds_b128
// (ASYNCcnt); relu intermediate re-laid-out C->A through a per-wave LDS tile.
// out may alias agg (each wave reads its rows before writing them).
// ---------------------------------------------------------------------------
#define YSTR 68   // 16x64 tile with stride 68 floats: 4l+k bank pattern
__global__ __launch_bounds__(128) void gine_node_kernel(
    const float* __restrict__ xin,    // [N,64]
    const float* agg,                 // [N,64]  (no restrict: may alias out)
    const float* __restrict__ w1, const float* __restrict__ b1,
    const float* __restrict__ w2, const float* __restrict__ b2,
    float* out,                       // [N,64]
    int nNodes)
{
    __shared__ float sW1[DCH * DCH];
    __shared__ float sW2[DCH * DCH];
    __shared__ float sY[4][16 * YSTR];

    // Async-stage both 64x64 weight matrices: 128 lanes x 16B x 8 iters = 16KB each.
#pragma unroll
    for (int it = 0; it < 8; ++it) {
        const int fidx = (it * 128 + threadIdx.x) * 4;  // float index, 16B granules
        async_copy16(&sW1[fidx], w1 + fidx);
        async_copy16(&sW2[fidx], w2 + fidx);
    }
    wait_asynccnt0();
    __syncthreads();

    const int lane = threadIdx.x & 31;
    const int half = lane >> 4;
    const int l15  = lane & 15;
    const int wslot = threadIdx.x >> 5;
    float* Y = &sY[wslot][0];

    const int waveId = (blockIdx.x * (blockDim.x >> 5)) + wslot;
    const int nWaves = (gridDim.x * blockDim.x) >> 5;

    float bias1[4], bias2[4];
#pragma unroll
    for (int nt = 0; nt < 4; ++nt) {
        bias1[nt] = b1[nt * 16 + l15];
        bias2[nt] = b2[nt * 16 + l15];
    }

    const int nTiles = nNodes >> 4;   // N = 100000 -> 6250 tiles exactly
    for (int tile = waveId; tile < nTiles; tile += nWaves) {
        const int nbase = tile * 16;

        // A1 = (xin + agg) row nbase+l15; K=64 -> 16 K-steps, v2f per step.
        v2f A1[16];
        const float* xr = xin + (nbase + l15) * DCH + half * 2;
        const float* ar = agg + (nbase + l15) * DCH + half * 2;
#pragma unroll
        for (int ks = 0; ks < 16; ++ks) {
            v2f xa = *(const v2f*)(xr + ks * 4);
            v2f aa = *(const v2f*)(ar + ks * 4);
            A1[ks].x = xa.x + aa.x;
            A1[ks].y = xa.y + aa.y;
        }

        // GEMM1 + relu -> LDS tile in A-layout-friendly form
#pragma unroll
        for (int nt = 0; nt < 4; ++nt) {
            v8f c;
#pragma unroll
            for (int i = 0; i < 8; ++i) c[i] = bias1[nt];
#pragma unroll
            for (int ks = 0; ks < 16; ++ks) {
                const int k = ks * 4 + half * 2;
                v2f b = *(const v2f*)(sW1 + (nt * 16 + l15) * DCH + k);
                c = __builtin_amdgcn_wmma_f32_16x16x4_f32(
                        false, A1[ks], false, b, (short)0, c, false, false);
            }
            const int col = nt * 16 + l15;
#pragma unroll
            for (int r = 0; r < 8; ++r) {
                float v = c[r];
                Y[(r + half * 8) * YSTR + col] = v > 0.f ? v : 0.f;
            }
        }

        // GEMM2: A from LDS tile, B from sW2, write h_out
#pragma unroll
        for (int nt = 0; nt < 4; ++nt) {
            v8f c;
#pragma unroll
            for (int i = 0; i < 8; ++i) c[i] = bias2[nt];
#pragma unroll
            for (int ks = 0; ks < 16; ++ks) {
                const int k = ks * 4 + half * 2;
                v2f a = *(const v2f*)(Y + l15 * YSTR + k);
                v2f b = *(const v2f*)(sW2 + (nt * 16 + l15) * DCH + k);
                c = __builtin_amdgcn_wmma_f32_16x16x4_f32(
                        false, a, false, b, (short)0, c, false, false);
            }
            const int col = nt * 16 + l15;
#pragma unroll
            for (int r = 0; r < 8; ++r)
                out[(nbase + r + half * 8) * DCH + col] = c[r];
        }
    }
}

// ---------------------------------------------------------------------------
// Mean-pool accumulate: atomic sums per graph + node counts.
// ---------------------------------------------------------------------------
__global__ __launch_bounds__(256) void pool_accum_kernel(
    const float* __restrict__ h, const int* __restrict__ batch,
    float* __restrict__ sums, float* __restrict__ cnt, int nNodes)
{
    const int i = blockIdx.x * blockDim.x + threadIdx.x;
    if (i >= nNodes * DCH) return;
    const int node = i >> 6;
    const int c = i & 63;
    const int b = batch[node];
    atomicAdd(&sums[b * DCH + c], h[i]);
    if (c == 0) atomicAdd(&cnt[b], 1.0f);
}

// ---------------------------------------------------------------------------
// Readout: one wave per graph; pooled = sums/max(cnt,1); out = pooled @ lw.T + lb
// ---------------------------------------------------------------------------
__global__ __launch_bounds__(256) void readout_kernel(
    const float* __restrict__ sums, const float* __restrict__ cnt,
    const float* __restrict__ lw, const float* __restrict__ lb,
    float* __restrict__ out, int nGraphs)
{
    const int wave = (blockIdx.x * blockDim.x + threadIdx.x) >> 5;
    const int lane = threadIdx.x & 31;
    if (wave >= nGraphs) return;
    const float inv = 1.0f / fmaxf(cnt[wave], 1.0f);
    const float p0 = sums[wave * DCH + lane] * inv;
    const float p1 = sums[wave * DCH + 32 + lane] * inv;
    float acc = p0 * lw[lane] + p1 * lw[32 + lane];
#pragma unroll
    for (int off = 16; off > 0; off >>= 1) acc += __shfl_down(acc, off, 32);
    if (lane == 0) out[wave] = acc + lb[0];
}

// ---------------------------------------------------------------------------
extern "C" void kernel_launch(void* const* d_in, const int* in_sizes, int n_in,
                              void* d_out, int out_size, void* d_ws, size_t ws_size,
                              hipStream_t stream) {
    const float* x      = (const float*)d_in[0];
    const int*   eidx   = (const int*)  d_in[1];
    const int*   batch  = (const int*)  d_in[2];
    const float* eattr  = (const float*)d_in[3];
    const float* elin1w = (const float*)d_in[4];
    const float* elin1b = (const float*)d_in[5];
    const float* m1w1   = (const float*)d_in[6];
    const float* m1b1   = (const float*)d_in[7];
    const float* m1w2   = (const float*)d_in[8];
    const float* m1b2   = (const float*)d_in[9];
    const float* elin2w = (const float*)d_in[10];
    const float* elin2b = (const float*)d_in[11];
    const float* m2w1   = (const float*)d_in[12];
    const float* m2b1   = (const float*)d_in[13];
    const float* m2w2   = (const float*)d_in[14];
    const float* m2b2   = (const float*)d_in[15];
    const float* linw   = (const float*)d_in[16];
    const float* linb   = (const float*)d_in[17];
    float* out = (float*)d_out;

    const int N = in_sizes[0] / DCH;       // 100000
    const int E = in_sizes[1] / 2;         // 1600000
    const int G = out_size;                // 512
    const int* src = eidx;
    const int* dst = eidx + E;

    float* agg  = (float*)d_ws;            // [N,64]; also reused as h2
    float* h1   = agg + (size_t)N * DCH;   // [N,64]
    float* sums = h1 + (size_t)N * DCH;    // [G,64]
    float* cnt  = sums + (size_t)G * DCH;  // [G]

    const int edgeBlocks = 512;            // 4096 waves, ~24 edge-tiles each
    const int nodeBlocks = 512;            // 2048 waves over 6250 node-tiles
    const int poolBlocks = (N * DCH + 255) / 256;
    const int roBlocks   = (G + 7) / 8;    // 8 waves per 256-thread block

    // ---- layer 1 ----
    hipMemsetAsync(agg, 0, (size_t)N * DCH * sizeof(float), stream);
    gine_edge_kernel<<<edgeBlocks, 256, 0, stream>>>(x, eattr, src, dst,
                                                     elin1w, elin1b, agg, E);
    gine_node_kernel<<<nodeBlocks, 128, 0, stream>>>(x, agg, m1w1, m1b1,
                                                     m1w2, m1b2, h1, N);
    // ---- layer 2 ----
    hipMemsetAsync(agg, 0, (size_t)N * DCH * sizeof(float), stream);
    gine_edge_kernel<<<edgeBlocks, 256, 0, stream>>>(h1, eattr, src, dst,
                                                     elin2w, elin2b, agg, E);
    gine_node_kernel<<<nodeBlocks, 128, 0, stream>>>(h1, agg, m2w1, m2b1,
                                                     m2w2, m2b2, /*out=*/agg, N);
    // ---- pool + readout ----
    hipMemsetAsync(sums, 0, (size_t)(G * DCH + G) * sizeof(float), stream);
    pool_accum_kernel<<<poolBlocks, 256, 0, stream>>>(agg, batch, sums, cnt, N);
    readout_kernel<<<roBlocks, 256, 0, stream>>>(sums, cnt, linw, linb, out, G);
}